// LookupTableLayer_21706764714522
// MI455X (gfx1250) — compile-verified
//
#include <hip/hip_runtime.h>
#include <math.h>

#define ADD_RATE 0.1f
#define D 128
#define RED_BLOCKS 64
#define RED_THREADS 256

// Native clang vector type: accepted by __builtin_nontemporal_* and lowers
// to b128 global memory ops.
typedef float v4f __attribute__((ext_vector_type(4)));

// ---------------------------------------------------------------------------
// Kernel 1: per-block max over one table (blockIdx.y selects table_x/table_y)
// ---------------------------------------------------------------------------
__global__ void lut_max_partial(const float* __restrict__ tx,
                                const float* __restrict__ ty,
                                float* __restrict__ partial, int n4) {
    const v4f* src = (const v4f*)(blockIdx.y == 0 ? tx : ty);
    float m = -INFINITY;
    int stride = gridDim.x * blockDim.x;
    for (int i = blockIdx.x * blockDim.x + threadIdx.x; i < n4; i += stride) {
        v4f v = src[i];
        m = fmaxf(m, fmaxf(fmaxf(v.x, v.y), fmaxf(v.z, v.w)));
    }
    __shared__ float s[RED_THREADS];
    s[threadIdx.x] = m;
    __syncthreads();
    for (int off = RED_THREADS / 2; off > 0; off >>= 1) {
        if (threadIdx.x < off)
            s[threadIdx.x] = fmaxf(s[threadIdx.x], s[threadIdx.x + off]);
        __syncthreads();
    }
    if (threadIdx.x == 0)
        partial[blockIdx.y * RED_BLOCKS + blockIdx.x] = s[0];
}

// ---------------------------------------------------------------------------
// Kernel 2: reduce the partials, emit scales[] = ADD_RATE / max
// ---------------------------------------------------------------------------
__global__ void lut_max_final(const float* __restrict__ partial,
                              float* __restrict__ scales) {
    __shared__ float s0[RED_BLOCKS];
    __shared__ float s1[RED_BLOCKS];
    int t = threadIdx.x;
    s0[t] = partial[t];
    s1[t] = partial[RED_BLOCKS + t];
    __syncthreads();
    for (int off = RED_BLOCKS / 2; off > 0; off >>= 1) {
        if (t < off) {
            s0[t] = fmaxf(s0[t], s0[t + off]);
            s1[t] = fmaxf(s1[t], s1[t + off]);
        }
        __syncthreads();
    }
    if (t == 0) {
        scales[0] = ADD_RATE / s0[0];
        scales[1] = ADD_RATE / s1[0];
    }
}

// ---------------------------------------------------------------------------
// Kernel 3: precompute pair-summed tables
//   p[i,k] = s*(t[i,2k]+t[i,2k+1]) + (f[i,2k]+f[i,2k+1]),  k in [0,64)
// One v4f of output per thread (reads 2x v4f table + 2x v4f fixed).
// blockIdx.y selects (table_x -> px) vs (table_y -> py).
// px/py are written with default (RT) temporal hint: they must stay in L2
// for the random gather that follows.
// ---------------------------------------------------------------------------
__global__ void lut_build(const float* __restrict__ tx,
                          const float* __restrict__ ty,
                          const float* __restrict__ fixedt,
                          const float* __restrict__ scales,
                          float* __restrict__ px,
                          float* __restrict__ py, int table_len) {
    int e = blockIdx.x * blockDim.x + threadIdx.x;
    int total = table_len * 16;                 // 16 v4f per output row
    if (e >= total) return;
    int i = e >> 4;
    int c = e & 15;
    int sel = blockIdx.y;
    float s = scales[sel];
    const float* table = sel ? ty : tx;
    float* pt = sel ? py : px;

    const v4f* t = (const v4f*)(table + (size_t)i * D + c * 8);
    const v4f* f = (const v4f*)(fixedt + (size_t)i * D + c * 8);
    v4f t0 = t[0], t1 = t[1];
    v4f f0 = f[0], f1 = f[1];
    v4f o;
    o.x = s * (t0.x + t0.y) + (f0.x + f0.y);
    o.y = s * (t0.z + t0.w) + (f0.z + f0.w);
    o.z = s * (t1.x + t1.y) + (f1.x + f1.y);
    o.w = s * (t1.z + t1.w) + (f1.z + f1.w);
    ((v4f*)(pt + (size_t)i * 64))[c] = o;
}

// ---------------------------------------------------------------------------
// Kernel 4: gather. One wave32 per output row of 128 floats.
//   lanes 0..15  : v4f chunks of px[ix]   -> out[row][0:64]
//   lanes 16..31 : v4f chunks of py[iy]   -> out[row][64:128]
// Output stores are NON-TEMPORAL: 134 MB written once, never re-read;
// NT keeps L2 reserved for the 2.1 MB px/py gather tables.
// ---------------------------------------------------------------------------
__global__ void lut_gather(const int* __restrict__ pos,
                           const float* __restrict__ px,
                           const float* __restrict__ py,
                           float* __restrict__ out, int rows) {
    int lane = threadIdx.x & 31;
    int wave = threadIdx.x >> 5;
    int r = blockIdx.x * 8 + wave;
    if (r >= rows) return;
    int half = lane >> 4;
    int c = lane & 15;
    int idx = __builtin_nontemporal_load(pos + 2 * r + half);  // single-touch
    if (idx < 0) idx = 1;
    const float* src = half ? py : px;
    v4f v = ((const v4f*)(src + (size_t)idx * 64))[c];         // RT: L2-hot
    __builtin_nontemporal_store(v, (v4f*)(out + (size_t)r * D) + lane);
}

// ---------------------------------------------------------------------------
// Fallback (if workspace too small for px/py): direct fused gather.
// ---------------------------------------------------------------------------
__global__ void lut_gather_direct(const int* __restrict__ pos,
                                  const float* __restrict__ tx,
                                  const float* __restrict__ ty,
                                  const float* __restrict__ fixedt,
                                  const float* __restrict__ scales,
                                  float* __restrict__ out, int rows) {
    int lane = threadIdx.x & 31;
    int wave = threadIdx.x >> 5;
    int r = blockIdx.x * 8 + wave;
    if (r >= rows) return;
    int half = lane >> 4;
    int c = lane & 15;
    int idx = __builtin_nontemporal_load(pos + 2 * r + half);
    if (idx < 0) idx = 1;
    float s = scales[half];
    const float* t = (half ? ty : tx) + (size_t)idx * D + c * 8;
    const float* f = fixedt + (size_t)idx * D + c * 8;
    v4f t0 = ((const v4f*)t)[0], t1 = ((const v4f*)t)[1];
    v4f f0 = ((const v4f*)f)[0], f1 = ((const v4f*)f)[1];
    v4f o;
    o.x = s * (t0.x + t0.y) + (f0.x + f0.y);
    o.y = s * (t0.z + t0.w) + (f0.z + f0.w);
    o.z = s * (t1.x + t1.y) + (f1.x + f1.y);
    o.w = s * (t1.z + t1.w) + (f1.z + f1.w);
    __builtin_nontemporal_store(o, (v4f*)(out + (size_t)r * D) + lane);
}

extern "C" void kernel_launch(void* const* d_in, const int* in_sizes, int n_in,
                              void* d_out, int out_size, void* d_ws, size_t ws_size,
                              hipStream_t stream) {
    const int*   positions = (const int*)d_in[0];
    const float* fixedt    = (const float*)d_in[1];
    const float* tx        = (const float*)d_in[2];
    const float* ty        = (const float*)d_in[3];
    float*       out       = (float*)d_out;

    const int table_elems = in_sizes[1];        // table_len * D
    const int table_len   = table_elems / D;
    const int rows        = in_sizes[0] / 2;    // B*M*R
    const int n4          = table_elems / 4;    // v4f count per table

    // Workspace layout:
    //   [0, 512)            : 128 partial maxima
    //   [512, 520)          : scales[2]
    //   [1024, 1024+S)      : px  (table_len * 64 floats)
    //   [1024+S, 1024+2S)   : py
    float* partial = (float*)d_ws;
    float* scales  = partial + 2 * RED_BLOCKS;
    float* px      = (float*)((char*)d_ws + 1024);
    float* py      = px + (size_t)table_len * 64;
    const size_t needed = 1024 + 2ull * (size_t)table_len * 64 * sizeof(float);

    // 1) partial max per table
    dim3 g1(RED_BLOCKS, 2);
    lut_max_partial<<<g1, RED_THREADS, 0, stream>>>(tx, ty, partial, n4);

    // 2) finalize scales
    lut_max_final<<<1, RED_BLOCKS, 0, stream>>>(partial, scales);

    if (ws_size >= needed) {
        // 3) build pair-summed tables (L2-resident, ~2.1 MB)
        int total = table_len * 16;
        dim3 g3((total + 255) / 256, 2);
        lut_build<<<g3, 256, 0, stream>>>(tx, ty, fixedt, scales, px, py, table_len);

        // 4) gather: one wave per output row, NT stores
        lut_gather<<<(rows + 7) / 8, 256, 0, stream>>>(positions, px, py, out, rows);
    } else {
        // Fallback: fused direct gather
        lut_gather_direct<<<(rows + 7) / 8, 256, 0, stream>>>(
            positions, tx, ty, fixedt, scales, out, rows);
    }
}